// SimilarityBasedFilter_48009144434939
// MI455X (gfx1250) — compile-verified
//
#include <hip/hip_runtime.h>
#include <stdint.h>

// Bilateral-style 5x5 depth filter, (B,1,512,512) fp32.
// Memory-bound (16.8 MB total traffic -> ~0.72us floor at 23.3 TB/s), so:
//  - skip the dead normal/covariance computation entirely
//  - stage each input tile into LDS exactly once via gfx1250 async
//    global->LDS copies (ASYNCcnt path), zero-padded halo
//  - register sliding window: 5 LDS reads/pixel instead of 25
//  - one v_exp_f32 per tap via exp2

#define H 512
#define W 512
#define TILE_X 64
#define TILE_Y 32
#define PADR 2
#define TW (TILE_X + 2 * PADR)      // 68
#define TH (TILE_Y + 2 * PADR)      // 36
#define TILE_ELEMS (TW * TH)        // 2448
#define THREADS 256
// exp(-diff^2 / (2*0.1^2)) = exp2(-diff^2 * 50 * log2(e))
#define NEG_SCALE_LOG2E (-72.13475204444817f)

__global__ __launch_bounds__(THREADS)
void SimilarityBasedFilter_48009144434939_kernel(const float* __restrict__ in,
                                                 float* __restrict__ out) {
    __shared__ float tile[TILE_ELEMS];   // only LDS object -> starts at LDS offset 0

    const int tid = threadIdx.x;
    const int tx0 = blockIdx.x * TILE_X;
    const int ty0 = blockIdx.y * TILE_Y;
    const int b   = blockIdx.z;
    const float* src = in + (size_t)b * (H * W);

    // ---- Stage halo tile into LDS with async global->LDS (zero padding OOB) ----
    #pragma unroll
    for (int i = 0; i < (TILE_ELEMS + THREADS - 1) / THREADS; ++i) {
        int e = i * THREADS + tid;
        if (e < TILE_ELEMS) {
            int ly = e / TW;
            int lx = e - ly * TW;
            int gy = ty0 - PADR + ly;
            int gx = tx0 - PADR + lx;
            if ((unsigned)gy < (unsigned)H && (unsigned)gx < (unsigned)W) {
                uint32_t lds_addr = (uint32_t)(e * 4);   // byte address inside LDS alloc
                const float* gp = src + (gy * W + gx);
                asm volatile("global_load_async_to_lds_b32 %0, %1, off"
                             :: "v"(lds_addr), "v"(gp)
                             : "memory");
            } else {
                tile[e] = 0.0f;    // matches jnp.pad zero padding
            }
        }
    }
    asm volatile("s_wait_asynccnt 0x0" ::: "memory");   // our wave's async copies done
    __syncthreads();                                    // everyone's copies visible

    // ---- Compute: each thread does 8 consecutive rows, 5x5 register window ----
    const int lx = tid & (TILE_X - 1);         // 0..63  (contiguous -> coalesced stores)
    const int ry = (tid >> 6) * 8;             // 0,8,16,24

    float win[5][5];
    #pragma unroll
    for (int i = 0; i < 4; ++i)
        #pragma unroll
        for (int j = 0; j < 5; ++j)
            win[i][j] = tile[(ry + i) * TW + lx + j];

    #pragma unroll
    for (int p = 0; p < 8; ++p) {
        // slide: bring in the new bottom row (5 LDS reads per pixel)
        #pragma unroll
        for (int j = 0; j < 5; ++j)
            win[(p + 4) % 5][j] = tile[(ry + p + 4) * TW + lx + j];

        const float dc = win[(p + 2) % 5][2];  // center tap
        float s = 0.0f, t = 0.0f;
        #pragma unroll
        for (int i = 0; i < 5; ++i) {
            #pragma unroll
            for (int j = 0; j < 5; ++j) {
                float d = win[(p + i) % 5][j];
                float diff = d - dc;
                float w = __builtin_amdgcn_exp2f(diff * diff * NEG_SCALE_LOG2E);
                s = fmaf(d, w, s);
                t += w;
            }
        }
        const int gy = ty0 + ry + p;
        out[(size_t)b * (H * W) + gy * W + tx0 + lx] = s / (t + 1e-6f);
    }
}

extern "C" void kernel_launch(void* const* d_in, const int* in_sizes, int n_in,
                              void* d_out, int out_size, void* d_ws, size_t ws_size,
                              hipStream_t stream) {
    (void)n_in; (void)d_ws; (void)ws_size; (void)out_size;
    const float* depth = (const float*)d_in[0];
    float* out = (float*)d_out;
    const int batches = in_sizes[0] / (H * W);   // 8 for the reference shapes

    dim3 block(THREADS, 1, 1);
    dim3 grid(W / TILE_X, H / TILE_Y, batches);
    SimilarityBasedFilter_48009144434939_kernel<<<grid, block, 0, stream>>>(depth, out);
}